// WrappingLoss_9861244912360
// MI455X (gfx1250) — compile-verified
//
#include <hip/hip_runtime.h>
#include <hip/hip_bf16.h>

typedef float v2f __attribute__((ext_vector_type(2)));
typedef float v8f __attribute__((ext_vector_type(8)));

// ---------------------------------------------------------------------------
// Prep: pts [n,3] row-major ->
//   aug[i]  = {x, y, z, x^2+y^2+z^2}                    (A / C side, i < n)
//   blay[.] = B operand pre-swizzled into the exact WMMA register layout:
//             tile t, lane l<16  : blay[t*32+l]    = {x, y}   of point t*16+l
//             tile t, lane l>=16 : blay[t*32+l]    = {z, b2}  of point t*16+(l-16)
//   One extra padding tile (duplicating the last point) allows the main
//   loop to prefetch one tile ahead with no bounds check.
// ---------------------------------------------------------------------------
__global__ void __launch_bounds__(256)
chamfer_prep_kernel(const float* __restrict__ pts, float4* __restrict__ aug,
                    v2f* __restrict__ blay, int n, int nPad) {
    int i = blockIdx.x * blockDim.x + threadIdx.x;
    if (i >= nPad) return;
    int s = i < n ? i : n - 1;               // padding duplicates last point
    float x = pts[3 * s + 0];
    float y = pts[3 * s + 1];
    float z = pts[3 * s + 2];
    float w = x * x + y * y + z * z;
    if (i < n) aug[i] = make_float4(x, y, z, w);
    int tile = i >> 4, r = i & 15;
    v2f lo; lo[0] = x; lo[1] = y;
    v2f hi; hi[0] = z; hi[1] = w;
    blay[tile * 32 + r]      = lo;           // lanes 0-15 : K0,K1
    blay[tile * 32 + 16 + r] = hi;           // lanes 16-31: K2,K3
}

// ---------------------------------------------------------------------------
// Each wave owns a strip of 16 rows; scans all column tiles with
// V_WMMA_F32_16X16X4_F32 computing D = A*B + C where
//   A row m = [-2x, -2y, -2z, 1], B col n = [x, y, z, ||b||^2],
//   C[m][n] = ||a_m||^2  =>  D[m][n] = squared distance.
// B tiles are loaded pre-swizzled (one b64 per lane per tile), pipelined
// one tile ahead via the padding tile. The >=0 clamp commutes with min and
// is applied once at the end. shfl_xor min-tree gives 16 row-mins per wave.
// ---------------------------------------------------------------------------
__global__ void __launch_bounds__(256)
chamfer_rowmin_kernel(const float4* __restrict__ rowsAug,
                      const v2f* __restrict__ colsLay,
                      float* __restrict__ rowMin,
                      int nRows, int nColTiles) {
    const int lane        = threadIdx.x & 31;
    const int waveInBlock = threadIdx.x >> 5;
    const int strip       = blockIdx.x * (blockDim.x >> 5) + waveInBlock;
    const int rowBase     = strip * 16;
    if (rowBase >= nRows) return;

    const int half = (lane >> 4) & 1;   // 0: lanes 0-15 (K0/K1, M0-7), 1: lanes 16-31 (K2/K3, M8-15)
    const int l16  = lane & 15;

    // ---- A matrix (16x4), loop invariant ----
    int ra = rowBase + l16;
    if (ra >= nRows) ra = nRows - 1;           // clamp (harmless for mins)
    float4 ap = rowsAug[ra];
    v2f a;
    if (half == 0) { a[0] = -2.0f * ap.x; a[1] = -2.0f * ap.y; }
    else           { a[0] = -2.0f * ap.z; a[1] = 1.0f;         }

    // ---- C matrix: C[m][n] = ||a_m||^2 ; VGPR k holds row M = k + 8*half ----
    v8f c;
#pragma unroll
    for (int k = 0; k < 8; ++k) {
        int rc = rowBase + half * 8 + k;
        if (rc >= nRows) rc = nRows - 1;
        c[k] = rowsAug[rc].w;
    }

    v8f rmin;
#pragma unroll
    for (int k = 0; k < 8; ++k) rmin[k] = 3.402823466e+38f;

    // ---- main loop: 1 b64 load (next tile) + 1 WMMA + 8 mins per tile ----
    int idx = lane;
    v2f b = colsLay[idx];
#pragma unroll 4
    for (int t = 0; t < nColTiles; ++t) {
        idx += 32;
        v2f bn = colsLay[idx];             // padding tile makes this safe at t末
        v8f d = __builtin_amdgcn_wmma_f32_16x16x4_f32(
            /*neg_a=*/false, a, /*neg_b=*/false, b,
            /*c_mod=*/(short)0, c, /*reuse_a=*/false, /*reuse_b=*/false);
#pragma unroll
        for (int k = 0; k < 8; ++k)
            rmin[k] = fminf(rmin[k], d[k]);
        b = bn;
    }

    // ---- min-reduce across the 16 lanes of each half (columns) ----
#pragma unroll
    for (int off = 1; off <= 8; off <<= 1) {
#pragma unroll
        for (int k = 0; k < 8; ++k) {
            float o = __shfl_xor(rmin[k], off, 32);
            rmin[k] = fminf(rmin[k], o);
        }
    }

    if (l16 == 0) {
#pragma unroll
        for (int k = 0; k < 8; ++k) {
            int r = rowBase + half * 8 + k;
            if (r < nRows) rowMin[r] = fmaxf(rmin[k], 0.0f);  // clamp hoisted here
        }
    }
}

// ---------------------------------------------------------------------------
// Deterministic final reduction: out = mean(rmA) + mean(rmB)
// Single block, fixed-order grid-stride accumulation + LDS tree.
// ---------------------------------------------------------------------------
__global__ void __launch_bounds__(256)
chamfer_finalize_kernel(const float* __restrict__ rmA, int n,
                        const float* __restrict__ rmB, int m,
                        float* __restrict__ out) {
    __shared__ float smA[256];
    __shared__ float smB[256];
    float sA = 0.0f, sB = 0.0f;
    for (int i = threadIdx.x; i < n; i += 256) sA += rmA[i];
    for (int i = threadIdx.x; i < m; i += 256) sB += rmB[i];
    smA[threadIdx.x] = sA;
    smB[threadIdx.x] = sB;
    __syncthreads();
    for (int off = 128; off > 0; off >>= 1) {
        if (threadIdx.x < off) {
            smA[threadIdx.x] += smA[threadIdx.x + off];
            smB[threadIdx.x] += smB[threadIdx.x + off];
        }
        __syncthreads();
    }
    if (threadIdx.x == 0)
        out[0] = smA[0] / (float)n + smB[0] / (float)m;
}

// ---------------------------------------------------------------------------
extern "C" void kernel_launch(void* const* d_in, const int* in_sizes, int n_in,
                              void* d_out, int out_size, void* d_ws, size_t ws_size,
                              hipStream_t stream) {
    const float* p_hat = (const float*)d_in[0];   // [n, 3]
    const float* p     = (const float*)d_in[1];   // [m, 3]
    const int n = in_sizes[0] / 3;
    const int m = in_sizes[1] / 3;

    const int tA = (n + 15) / 16;                 // column tiles when cloud A is B-side
    const int tB = (m + 15) / 16;
    const int padA = (tA + 1) * 16;               // +1 padding tile for pipelining
    const int padB = (tB + 1) * 16;

    float*  ws    = (float*)d_ws;
    float4* augA  = (float4*)ws;                                  // 4n floats
    float4* augB  = (float4*)(ws + 4 * (size_t)n);                // 4m floats
    v2f*    blayA = (v2f*)(ws + 4 * (size_t)(n + m));             // padA*2*... (2 floats per entry, 2 entries per point)
    v2f*    blayB = blayA + 2 * (size_t)padA;
    float*  rmA   = (float*)(blayB + 2 * (size_t)padB);           // n floats (min over axis 1)
    float*  rmB   = rmA + n;                                      // m floats (min over axis 0)

    chamfer_prep_kernel<<<(padA + 255) / 256, 256, 0, stream>>>(p_hat, augA, blayA, n, padA);
    chamfer_prep_kernel<<<(padB + 255) / 256, 256, 0, stream>>>(p, augB, blayB, m, padB);

    // 256 threads = 8 wave32s; one 16-row strip per wave
    chamfer_rowmin_kernel<<<(tA + 7) / 8, 256, 0, stream>>>(augA, blayB, rmA, n, tB);
    chamfer_rowmin_kernel<<<(tB + 7) / 8, 256, 0, stream>>>(augB, blayA, rmB, m, tA);

    chamfer_finalize_kernel<<<1, 256, 0, stream>>>(rmA, n, rmB, m, (float*)d_out);
}